// HingeLoss_27367531610488
// MI455X (gfx1250) — compile-verified
//
#include <hip/hip_runtime.h>

typedef __attribute__((ext_vector_type(4))) float v4f;
typedef __attribute__((ext_vector_type(2))) float v2f;
typedef __attribute__((ext_vector_type(8))) float v8f;

constexpr int NBLK = 2048;   // stage-1 blocks -> 2048 partials (32 chunks of 64 for WMMA)
constexpr int TPB  = 256;    // 8 wave32 waves per block

__device__ __forceinline__ float relu4sum(v4f x) {
  float a = fmaxf(1.0f + x.x, 0.0f) + fmaxf(1.0f + x.y, 0.0f);
  float b = fmaxf(1.0f + x.z, 0.0f) + fmaxf(1.0f + x.w, 0.0f);
  return a + b;
}

// Stage 1: stream the whole matrix with non-temporal b128 loads, accumulate
// relu(1+x); fold in the per-row target corrections; one partial per block.
__global__ void __launch_bounds__(TPB) hinge_partials_kernel(
    const float* __restrict__ out, const int* __restrict__ tgt,
    float* __restrict__ partials, int B, int C) {
  const long long N  = (long long)B * (long long)C;
  const long long N4 = N >> 2;
  const v4f* p = (const v4f*)out;

  long long i = (long long)blockIdx.x * TPB + threadIdx.x;
  const long long stride = (long long)NBLK * TPB;

  float acc = 0.0f;
  // 4 independent 16B NT loads in flight per iteration (64B/thread/iter)
  for (; i + 3 * stride < N4; i += 4 * stride) {
    v4f x0 = __builtin_nontemporal_load(p + i);
    v4f x1 = __builtin_nontemporal_load(p + i + stride);
    v4f x2 = __builtin_nontemporal_load(p + i + 2 * stride);
    v4f x3 = __builtin_nontemporal_load(p + i + 3 * stride);
    acc += relu4sum(x0) + relu4sum(x1);
    acc += relu4sum(x2) + relu4sum(x3);
  }
  for (; i < N4; i += stride)
    acc += relu4sum(__builtin_nontemporal_load(p + i));
  // scalar remainder (N % 4), if any
  if (blockIdx.x == 0 && threadIdx.x == 0)
    for (long long j = (N4 << 2); j < N; ++j)
      acc += fmaxf(1.0f + out[j], 0.0f);

  // wave32 butterfly reduction (gfx1250: warpSize == 32)
  #pragma unroll
  for (int off = 16; off > 0; off >>= 1)
    acc += __shfl_xor(acc, off, 32);

  __shared__ float wsum[TPB / 32];
  if ((threadIdx.x & 31) == 0) wsum[threadIdx.x >> 5] = acc;
  __syncthreads();

  if (threadIdx.x == 0) {
    float s = 0.0f;
    #pragma unroll
    for (int w = 0; w < TPB / 32; ++w) s += wsum[w];

    // per-row corrections for this block's slice of rows:
    //   + relu(1 - c)           (term1 contribution)
    //   - relu(1 + c) / 10      (remove target column from the full term2 sum)
    const int rows_per_blk = (B + NBLK - 1) / NBLK;
    float corr = 0.0f;
    for (int r = 0; r < rows_per_blk; ++r) {
      long long row = (long long)blockIdx.x * rows_per_blk + r;
      if (row < (long long)B) {
        int t = tgt[row];
        float c = out[row * (long long)C + t];
        corr += fmaxf(1.0f - c, 0.0f) - 0.1f * fmaxf(1.0f + c, 0.0f);
      }
    }
    partials[blockIdx.x] = 0.1f * s + corr;
  }
}

// Stage 2 (single wave32): reduce 2048 partials via V_WMMA_F32_16X16X4_F32.
// A = ones(16x4), B = 4x16 chunk of partials, C accumulates column sums:
//   D[m][n] = sum_k B[k][n] + C[m][n]  (identical across m)
// 32 chunks of 64 cover all 2048 partials; a 16-lane shuffle finishes.
__global__ void __launch_bounds__(32) hinge_final_kernel(
    const float* __restrict__ partials, float* __restrict__ dout, int B) {
  const int lane = threadIdx.x;
  const int n    = lane & 15;   // column within 4x16 B tile
  const int kh   = lane >> 4;   // K-half select (VGPR0: K=0/2, VGPR1: K=1/3)

  v8f c = {};
  v2f a; a.x = 1.0f; a.y = 1.0f;   // A = all-ones 16x4

  #pragma unroll 1
  for (int chunk = 0; chunk < NBLK / 64; ++chunk) {
    const float* base = partials + chunk * 64;   // B[k][n] = base[k*16 + n]
    v2f b;
    b.x = base[(2 * kh + 0) * 16 + n];
    b.y = base[(2 * kh + 1) * 16 + n];
    c = __builtin_amdgcn_wmma_f32_16x16x4_f32(
        /*neg_a=*/false, a, /*neg_b=*/false, b,
        /*c_mod=*/(short)0, c, /*reuse_a=*/false, /*reuse_b=*/false);
  }

  // every lane's VGPR0 cell holds the column sum for its n (rows identical);
  // reduce the 16 columns with a width-16 butterfly.
  float r = c[0];
  #pragma unroll
  for (int off = 8; off > 0; off >>= 1)
    r += __shfl_xor(r, off, 16);

  if (lane == 0) dout[0] = r / (float)B;
}

extern "C" void kernel_launch(void* const* d_in, const int* in_sizes, int n_in,
                              void* d_out, int out_size, void* d_ws, size_t ws_size,
                              hipStream_t stream) {
  const float* out = (const float*)d_in[0];
  const int*   tgt = (const int*)d_in[1];

  const int B = in_sizes[1];                       // 8192
  const int C = (int)((long long)in_sizes[0] / B); // 50257

  float* partials = (float*)d_ws;                  // 2048 floats = 8 KB

  hinge_partials_kernel<<<NBLK, TPB, 0, stream>>>(out, tgt, partials, B, C);
  hinge_final_kernel<<<1, 32, 0, stream>>>(partials, (float*)d_out, B);
}